// SegPostProcessor_22548578304873
// MI455X (gfx1250) — compile-verified
//
#include <hip/hip_runtime.h>
#include <hip/hip_bf16.h>

// ---------------- constants from the reference ----------------
#define BB   4
#define QQ   300
#define CC   80
#define KK   300
#define NN   (QQ*CC)      // 24000
#define HM   160
#define SS   512
// d_out layout (floats): labels[4*300] | boxes[4*300*4] | scores[4*300] | masks[4*300*512*512]
#define OFF_BOXES  1200
#define OFF_SCORES 6000
#define OFF_MASKS  7200

typedef float v2f  __attribute__((ext_vector_type(2)));
typedef float v8f  __attribute__((ext_vector_type(8)));
typedef unsigned int u32x4 __attribute__((ext_vector_type(4)));
typedef int   i32x8 __attribute__((ext_vector_type(8)));
typedef int   i32x4 __attribute__((ext_vector_type(4)));

// order-preserving key: larger float -> larger uint
__device__ __forceinline__ unsigned int fkey(float f) {
    unsigned int u = __float_as_uint(f);
    return (u & 0x80000000u) ? ~u : (u | 0x80000000u);
}

// bilinear tap weight: window row r vs (base tap rj, frac fj)
__device__ __forceinline__ float wtap(int r, int rj, float fj) {
    return (r == rj) ? (1.0f - fj) : ((r == rj + 1) ? fj : 0.0f);
}

// ======================================================================
// Kernel 1: per-batch exact top-300 (radix-select + counting rank),
// box convert, label/score emit, crop ints + query gather idx -> ws.
// ======================================================================
__global__ __launch_bounds__(1024)
void seg_topk_kernel(const float* __restrict__ logits,
                     const float* __restrict__ boxes_in,
                     float* __restrict__ out,
                     int* __restrict__ ws) {
    const int b   = blockIdx.x;
    const int tid = threadIdx.x;
    const int bs  = blockDim.x;
    const float* lg = logits + (size_t)b * NN;

    __shared__ unsigned int hist[256];
    __shared__ unsigned int s_sel, s_rem;
    __shared__ unsigned int gt_key[320];
    __shared__ int          gt_idx[320];
    __shared__ int          eq_idx[512];
    __shared__ int          s_ngt, s_neq;

    // ---- 4-pass MSB radix select for the 300th-largest key ----
    unsigned int pfx = 0;
    int remaining = KK;
    for (int pass = 0; pass < 4; ++pass) {
        const int shift = 24 - 8 * pass;
        for (int i = tid; i < 256; i += bs) hist[i] = 0u;
        __syncthreads();
        for (int i = tid; i < NN; i += bs) {
            unsigned int key = fkey(lg[i]);
            bool match = (pass == 0) || ((key >> (shift + 8)) == pfx);
            if (match) atomicAdd(&hist[(key >> shift) & 255u], 1u);
        }
        __syncthreads();
        if (tid == 0) {
            int acc = 0, sel = 0;
            for (int bin = 255; bin >= 0; --bin) {
                int c = (int)hist[bin];
                if (acc + c >= remaining) { sel = bin; break; }
                acc += c;
            }
            s_sel = (unsigned int)sel;
            s_rem = (unsigned int)(remaining - acc);
        }
        __syncthreads();
        pfx = (pfx << 8) | s_sel;
        remaining = (int)s_rem;
        __syncthreads();
    }
    const unsigned int T = pfx;      // key of the 300th largest element
    const int need_eq = remaining;   // how many ==T to keep (lowest index first)

    // ---- compact survivors ----
    if (tid == 0) { s_ngt = 0; s_neq = 0; }
    __syncthreads();
    for (int i = tid; i < NN; i += bs) {
        unsigned int key = fkey(lg[i]);
        if (key > T) {
            int p = atomicAdd(&s_ngt, 1);
            if (p < 320) { gt_key[p] = key; gt_idx[p] = i; }
        } else if (key == T) {
            int p = atomicAdd(&s_neq, 1);
            if (p < 512) eq_idx[p] = i;
        }
    }
    __syncthreads();
    const int ngt = min(s_ngt, 320);
    const int neq = min(s_neq, 512);

    // ---- exact rank (sorted desc, ties by lower flat index) ----
    int rank = -1, myidx = -1;
    if (tid < ngt) {
        unsigned int mk = gt_key[tid]; int mi = gt_idx[tid];
        int r = 0;
        for (int j = 0; j < ngt; ++j) {
            unsigned int k2 = gt_key[j];
            if (k2 > mk || (k2 == mk && gt_idx[j] < mi)) r++;
        }
        rank = r; myidx = mi;
    } else if (tid >= 512 && (tid - 512) < neq) {
        int mi = eq_idx[tid - 512];
        int r = 0;
        for (int j = 0; j < neq; ++j) if (eq_idx[j] < mi) r++;
        if (r < need_eq) { rank = ngt + r; myidx = mi; }
    }

    if (rank >= 0 && rank < KK) {
        float l = lg[myidx];
        float score = 1.0f / (1.0f + expf(-l));
        int label = myidx % CC;
        int q     = myidx / CC;
        out[b * KK + rank]              = (float)label;
        out[OFF_SCORES + b * KK + rank] = score;
        const float* bx = boxes_in + ((size_t)b * QQ + q) * 4;
        float cx = bx[0], cy = bx[1], bw = bx[2], bh = bx[3];
        float x1 = (cx - 0.5f * bw) * (float)SS;
        float y1 = (cy - 0.5f * bh) * (float)SS;
        float x2 = (cx + 0.5f * bw) * (float)SS;
        float y2 = (cy + 0.5f * bh) * (float)SS;
        float* ob = out + OFF_BOXES + ((size_t)b * KK + rank) * 4;
        ob[0] = x1; ob[1] = y1; ob[2] = x2; ob[3] = y2;
        int xi1 = min(max((int)truncf(x1), 0), SS);
        int yi1 = min(max((int)truncf(y1), 0), SS);
        int xi2 = min(max((int)truncf(x2), 0), SS);
        int yi2 = min(max((int)truncf(y2), 0), SS);
        ws[b * KK + rank] = q;                       // gather index
        int4 cr; cr.x = xi1; cr.y = yi1; cr.z = xi2; cr.w = yi2;
        ((int4*)(ws + BB * KK))[b * KK + rank] = cr; // crop box
    }
}

// ======================================================================
// Kernel 2: fused gather + bilinear 160->512 (WMMA) + threshold + crop.
// grid = (1200, 8): blockIdx.x = b*300+k, blockIdx.y = 64-row strip.
// TDM stages the needed 23-row mask strip into LDS.
// ======================================================================
__global__ __launch_bounds__(256)
void seg_mask_kernel(const float* __restrict__ masks,
                     float* __restrict__ out,
                     const int* __restrict__ ws) {
    const int bk = blockIdx.x;          // b*300 + k
    const int gy = blockIdx.y;          // 0..7 -> y-tiles 4*gy .. 4*gy+3
    const int b  = bk / KK;
    const int q  = ws[bk];
    const int4 cr = ((const int4*)(ws + BB * KK))[bk];

    // source rows needed by this strip: 20*gy-1 .. 20*gy+21, clamped
    const int r0 = max(20 * gy - 1, 0);
    const int r1 = min(20 * gy + 21, HM - 1);
    const int nrows = r1 - r0 + 1;      // <= 23

    __shared__ float sm[23 * HM];       // 14.7 KB

    // ---- TDM async tensor load: strip of the gathered mask -> LDS ----
    if (threadIdx.x < 32) {
        const float* src = masks + ((size_t)b * QQ + q) * (HM * HM) + (size_t)r0 * HM;
        unsigned long long ga = (unsigned long long)(uintptr_t)src;
        unsigned int lds = (unsigned int)(uintptr_t)(void*)sm;

        u32x4 g0;
        g0[0] = 1u;                                            // count=1
        g0[1] = lds;                                           // lds_addr
        g0[2] = (unsigned int)(ga & 0xFFFFFFFFu);              // global_addr lo
        g0[3] = (unsigned int)((ga >> 32) & 0x01FFFFFFu) | (2u << 30); // hi | type=2

        i32x8 g1;
        g1[0] = (int)(2u << 16);          // data_size = 4B
        g1[1] = (int)(HM << 16);          // tensor_dim0 lo16 at bits[63:48]
        g1[2] = (int)((unsigned)nrows << 16); // dim0 hi=0 | tensor_dim1 lo16
        g1[3] = (int)(HM << 16);          // dim1 hi=0 | tile_dim0=160
        g1[4] = nrows;                    // tile_dim1 | tile_dim2=0
        g1[5] = HM;                       // tensor_dim0_stride lo32
        g1[6] = 0;                        // stride hi | dim1_stride lo
        g1[7] = 0;
        i32x4 z4; z4[0] = 0; z4[1] = 0; z4[2] = 0; z4[3] = 0;
        i32x8 z8; z8[0] = 0; z8[1] = 0; z8[2] = 0; z8[3] = 0;
        z8[4] = 0; z8[5] = 0; z8[6] = 0; z8[7] = 0;
        __builtin_amdgcn_tensor_load_to_lds(g0, g1, z4, z4, z8, 0);
        __builtin_amdgcn_s_wait_tensorcnt(0);
    }
    __syncthreads();

    // ---- per-lane constants (wave32) ----
    const int lane = threadIdx.x & 31;
    const int wave = threadIdx.x >> 5;
    const int jcol = lane & 15;
    const bool hi  = lane >= 16;
    const int k0   = hi ? 2 : 0;

    // bilinear tap for output index jcol within a 16-wide tile
    // (identical pattern every tile; window base = 5*t - 1)
    const float uj  = ((float)jcol + 0.5f) * 0.3125f - 0.5f;
    const float flu = floorf(uj);
    const float fj  = uj - flu;
    const int   rj  = (int)flu + 1;     // 0..5

    // Wx rows (B operand, K-major over lane halves); Wy cols (A operand) are identical
    v2f WB1; WB1.x = wtap(k0,     rj, fj); WB1.y = wtap(k0 + 1, rj, fj);
    v2f WB2; WB2.x = wtap(k0 + 4, rj, fj); WB2.y = wtap(k0 + 5, rj, fj);

    const v8f zero = {0.f,0.f,0.f,0.f,0.f,0.f,0.f,0.f};
    float* outmask = out + OFF_MASKS + (size_t)bk * (SS * SS);

    // 128 tiles per block (4 y-tiles x 32 x-tiles), 16 per wave
    for (int tile = wave; tile < 128; tile += 8) {
        const int rT = 4 * gy + (tile >> 5);
        const int tX = tile & 31;
        const int ybase = 5 * rT - 1;
        const int xbase = 5 * tX - 1;

        // A = clamped 8x8 input patch columns (rows striped over lanes)
        const int ysm = min(max(ybase + jcol, r0), r1) - r0;
        const float* rowp = sm + ysm * HM;
        v2f A1, A2;
        A1.x = rowp[min(max(xbase + k0,     0), HM - 1)];
        A1.y = rowp[min(max(xbase + k0 + 1, 0), HM - 1)];
        A2.x = rowp[min(max(xbase + k0 + 4, 0), HM - 1)];
        A2.y = rowp[min(max(xbase + k0 + 5, 0), HM - 1)];

        // step 1: T(16x16, rows 0..7 valid) = P(8x8) * Wx(8x16), K=8 as 2x K=4
        v8f Tm;
        Tm = __builtin_amdgcn_wmma_f32_16x16x4_f32(false, A2, false, WB2, (short)0, zero, false, false);
        Tm = __builtin_amdgcn_wmma_f32_16x16x4_f32(false, A1, false, WB1, (short)0, Tm,   false, false);

        // relayout T rows (D layout) into B operands for step 2
        float t2 = __shfl(Tm[2], jcol, 32);
        float t3 = __shfl(Tm[3], jcol, 32);
        float t6 = __shfl(Tm[6], jcol, 32);
        float t7 = __shfl(Tm[7], jcol, 32);
        v2f TB1; TB1.x = hi ? t2 : Tm[0]; TB1.y = hi ? t3 : Tm[1];
        v2f TB2; TB2.x = hi ? t6 : Tm[4]; TB2.y = hi ? t7 : Tm[5];

        // step 2: out(16x16) = Wy(16x8) * T(8x16)   (Wy cols == Wx rows pattern)
        v8f O;
        O = __builtin_amdgcn_wmma_f32_16x16x4_f32(false, WB2, false, TB2, (short)0, zero, false, false);
        O = __builtin_amdgcn_wmma_f32_16x16x4_f32(false, WB1, false, TB1, (short)0, O,    false, false);

        // threshold + crop + store (D layout: vgpr d -> row d / d+8)
        const int xo  = 16 * tX + jcol;
        const int yo0 = 16 * rT + (hi ? 8 : 0);
        const bool xok = (xo >= cr.x) & (xo < cr.z);
        float* op = outmask + (size_t)yo0 * SS + xo;
        #pragma unroll
        for (int d = 0; d < 8; ++d) {
            const int yo = yo0 + d;
            const bool ok = xok & (yo >= cr.y) & (yo < cr.w) & (O[d] > 0.0f);
            op[(size_t)d * SS] = ok ? 1.0f : 0.0f;
        }
    }
}

// ======================================================================
extern "C" void kernel_launch(void* const* d_in, const int* in_sizes, int n_in,
                              void* d_out, int out_size, void* d_ws, size_t ws_size,
                              hipStream_t stream) {
    const float* logits = (const float*)d_in[0];   // [4,300,80]
    const float* boxes  = (const float*)d_in[1];   // [4,300,4]
    const float* masks  = (const float*)d_in[2];   // [4,300,160,160]
    float* out = (float*)d_out;
    int*   ws  = (int*)d_ws;                       // qsel[1200] + int4 crop[1200]

    seg_topk_kernel<<<dim3(BB), dim3(1024), 0, stream>>>(logits, boxes, out, ws);
    seg_mask_kernel<<<dim3(BB * KK, 8), dim3(256), 0, stream>>>(masks, out, ws);
}